// Binarizer_77807627535051
// MI455X (gfx1250) — compile-verified
//
#include <hip/hip_runtime.h>

typedef float v2f __attribute__((ext_vector_type(2)));
typedef float v4f __attribute__((ext_vector_type(4)));
typedef float v8f __attribute__((ext_vector_type(8)));

// ---------------------------------------------------------------------------
// Kernel 0: zero the 256-bin count/sum workspace (harness does not re-poison;
// must be re-zeroed every call for graph-replay determinism).
// ---------------------------------------------------------------------------
__global__ void otsu_init_kernel(unsigned* __restrict__ gcnt, float* __restrict__ gsum) {
  const int i = threadIdx.x;           // 256 threads
  gcnt[i] = 0u;
  gsum[i] = 0.0f;
}

// ---------------------------------------------------------------------------
// Kernel 1: histogram (counts + per-bin value sums).
// Wave-private LDS sub-histograms (wave32 -> 8 waves/block) avoid inter-wave
// LDS atomic serialization; one global atomic per bin per block.
// Streaming b128 loads (regular-temporal: 128MB input fits in 192MB L2 and is
// re-read by the binarize pass). Bandwidth-bound.
// ---------------------------------------------------------------------------
__global__ void otsu_hist_kernel(const float* __restrict__ x, long long n,
                                 unsigned* __restrict__ gcnt, float* __restrict__ gsum) {
  __shared__ unsigned scnt[8 * 256];
  __shared__ float    ssum[8 * 256];
  const int tid = threadIdx.x;
  const int wid = tid >> 5;            // wave32 wave id within block
  unsigned* wcnt = &scnt[wid * 256];
  float*    wsum = &ssum[wid * 256];

  for (int i = tid; i < 8 * 256; i += blockDim.x) { scnt[i] = 0u; ssum[i] = 0.0f; }
  __syncthreads();

  const long long n4 = n >> 2;
  const v4f* __restrict__ x4 = reinterpret_cast<const v4f*>(x);
  const long long stride = (long long)gridDim.x * blockDim.x;
  for (long long i = (long long)blockIdx.x * blockDim.x + tid; i < n4; i += stride) {
    const v4f v = x4[i];
#pragma unroll
    for (int c = 0; c < 4; ++c) {
      const float f = v[c];
      int b = (int)floorf(f);
      b = b < 0 ? 0 : (b > 255 ? 255 : b);
      atomicAdd(&wcnt[b], 1u);
      atomicAdd(&wsum[b], f);
    }
  }
  // scalar tail (n not divisible by 4)
  if (blockIdx.x == 0) {
    for (long long i = (n4 << 2) + tid; i < n; i += blockDim.x) {
      const float f = x[i];
      int b = (int)floorf(f);
      b = b < 0 ? 0 : (b > 255 ? 255 : b);
      atomicAdd(&wcnt[b], 1u);
      atomicAdd(&wsum[b], f);
    }
  }
  __syncthreads();

  for (int i = tid; i < 256; i += blockDim.x) {
    unsigned c = 0; float s = 0.0f;
#pragma unroll
    for (int w = 0; w < 8; ++w) { c += scnt[w * 256 + i]; s += ssum[w * 256 + i]; }
    if (c)          atomicAdd(&gcnt[i], c);
    if (s != 0.0f)  atomicAdd(&gsum[i], s);
  }
}

// ---------------------------------------------------------------------------
// Kernel 2: single wave (32 threads). Computes the dual prefix-sum
// (ccount[t], csum[t]) for the 128 even thresholds t = 2j via chained
// V_WMMA_F32_16X16X4_F32:  D[m][n] = sum_k A[m][k] * B[k][n]
//   A[m][k] = [ bin(k) < t(m) ]   (triangular indicator, generated per lane)
//   B[k][0] = counts[bin(k)],  B[k][1] = sums[bin(k)],  other cols 0
// Layouts per CDNA5 ISA 7.12.2 (wave32):
//   A 16x4 f32: lanes 0-15 hold K={0,1} in VGPR{0,1}; lanes 16-31 hold K={2,3}
//   B 4x16 f32: lane = column; lane-half splits K the same way
//   D 16x16 f32: VGPR v, lanes 0-15 -> row v, lanes 16-31 -> row v+8
// B operand is built branch-free (uniform table offset + 0/1 scale) so the
// WMMA loop never manipulates EXEC (ISA: EXEC must be all 1s for WMMA).
// ---------------------------------------------------------------------------
__global__ void otsu_wmma_reduce_kernel(const unsigned* __restrict__ gcnt,
                                        const float* __restrict__ gsum,
                                        float* __restrict__ best_t,
                                        float n_total) {
  __shared__ float btab[512];  // [0,256) = counts (as f32), [256,512) = sums
  __shared__ float cc[128];    // ccount at t = 2j
  __shared__ float cs[128];    // csum   at t = 2j
  const int lane = threadIdx.x;        // 0..31, exactly one wave (EXEC all 1s)

  for (int i = lane; i < 256; i += 32) {
    btab[i]       = (float)gcnt[i];
    btab[256 + i] = gsum[i];
  }
  __syncthreads();

  const int   m      = lane & 15;              // A row within tile; B/D column
  const int   khalf  = lane >> 4;              // 0: K={0,1}, 1: K={2,3}
  const int   bsel   = (m == 1) ? 256 : 0;     // col 0 -> counts, col 1 -> sums
  const float bscale = (m < 2) ? 1.0f : 0.0f;  // cols 2..15 are zero

  for (int tile = 0; tile < 8; ++tile) {
    v8f acc = {};
    const int t = 2 * (tile * 16 + m);          // threshold for A row m
    for (int chunk = 0; chunk < 64; ++chunk) {
      const int kbase = chunk * 4 + khalf * 2;  // global bin of this lane's K pair
      v2f a, b;
      a[0] = (kbase + 0 < t) ? 1.0f : 0.0f;
      a[1] = (kbase + 1 < t) ? 1.0f : 0.0f;
      b[0] = btab[bsel + kbase + 0] * bscale;   // one ds_load_b64 + 2 muls
      b[1] = btab[bsel + kbase + 1] * bscale;
      acc = __builtin_amdgcn_wmma_f32_16x16x4_f32(false, a, false, b,
                                                  (short)0, acc, false, false);
    }
    // Scatter D columns 0 (ccount) and 1 (csum) to LDS.
    if (m < 2) {
#pragma unroll
      for (int v = 0; v < 8; ++v) {
        const int j = tile * 16 + v + khalf * 8;   // threshold index, t = 2j
        if (m == 0) cc[j] = acc[v];
        else        cs[j] = acc[v];
      }
    }
  }
  __syncthreads();

  // total_sum = sum of all per-bin sums (wave32 reduction)
  float ts = 0.0f;
  for (int i = lane; i < 256; i += 32) ts += btab[256 + i];
#pragma unroll
  for (int off = 16; off > 0; off >>= 1) ts += __shfl_xor(ts, off, 32);

  // Otsu between-class variance; first-occurrence argmax (matches jnp.argmax)
  float bestg = -1.0f; int bestj = 0;
  for (int j = lane; j < 128; j += 32) {
    const float c0 = cc[j], s0 = cs[j];
    const float c1 = n_total - c0, s1 = ts - s0;
    const float w0 = c0 / n_total,  w1 = c1 / n_total;
    const float u0 = (c0 > 0.0f) ? s0 / fmaxf(c0, 1.0f) : 0.0f;
    const float u1 = (c1 > 0.0f) ? s1 / fmaxf(c1, 1.0f) : 0.0f;
    const float d  = u0 - u1;
    const float g  = w0 * w1 * d * d;
    if (g > bestg) { bestg = g; bestj = j; }
  }
#pragma unroll
  for (int off = 16; off > 0; off >>= 1) {
    const float og = __shfl_xor(bestg, off, 32);
    const int   oj = __shfl_xor(bestj, off, 32);
    if (og > bestg || (og == bestg && oj < bestj)) { bestg = og; bestj = oj; }
  }
  if (lane == 0) *best_t = (float)(2 * bestj);
}

// ---------------------------------------------------------------------------
// Kernel 3: binarize. Loads of x are regular-temporal (likely L2-resident from
// the histogram pass: 128MB input vs 192MB L2); output uses non-temporal
// stores so the 128MB write stream does not evict x from L2.
// ---------------------------------------------------------------------------
__global__ void otsu_binarize_kernel(const float* __restrict__ x, float* __restrict__ out,
                                     const float* __restrict__ best_t, long long n) {
  const float t = *best_t;             // uniform scalar load
  const long long n4 = n >> 2;
  const v4f* __restrict__ x4 = reinterpret_cast<const v4f*>(x);
  v4f* __restrict__ o4 = reinterpret_cast<v4f*>(out);
  const long long stride = (long long)gridDim.x * blockDim.x;
  for (long long i = (long long)blockIdx.x * blockDim.x + threadIdx.x; i < n4; i += stride) {
    const v4f v = x4[i];
    v4f r;
    r[0] = (v[0] < t) ? 0.0f : 255.0f;
    r[1] = (v[1] < t) ? 0.0f : 255.0f;
    r[2] = (v[2] < t) ? 0.0f : 255.0f;
    r[3] = (v[3] < t) ? 0.0f : 255.0f;
    __builtin_nontemporal_store(r, &o4[i]);    // global_store_b128 TH=NT
  }
  if (blockIdx.x == 0) {
    for (long long i = (n4 << 2) + threadIdx.x; i < n; i += blockDim.x) {
      const float r = (x[i] < t) ? 0.0f : 255.0f;
      __builtin_nontemporal_store(r, &out[i]);
    }
  }
}

// ---------------------------------------------------------------------------
extern "C" void kernel_launch(void* const* d_in, const int* in_sizes, int n_in,
                              void* d_out, int out_size, void* d_ws, size_t ws_size,
                              hipStream_t stream) {
  const float* x = (const float*)d_in[0];
  float* out = (float*)d_out;
  const long long n = (long long)in_sizes[0];

  // d_ws layout: [0,256) u32 counts | [256,512) f32 sums | [512] f32 best_t
  unsigned* gcnt = (unsigned*)d_ws;
  float*    gsum = (float*)((char*)d_ws + 256 * sizeof(unsigned));
  float*    best = (float*)((char*)d_ws + 512 * sizeof(unsigned));

  otsu_init_kernel<<<1, 256, 0, stream>>>(gcnt, gsum);
  otsu_hist_kernel<<<1024, 256, 0, stream>>>(x, n, gcnt, gsum);
  otsu_wmma_reduce_kernel<<<1, 32, 0, stream>>>(gcnt, gsum, best, (float)n);
  otsu_binarize_kernel<<<8192, 256, 0, stream>>>(x, out, best, n);
}